// KnnModule_58385785422032
// MI455X (gfx1250) — compile-verified
//
#include <hip/hip_runtime.h>
#include <hip/hip_bf16.h>

typedef float v2f __attribute__((ext_vector_type(2)));
typedef float v8f __attribute__((ext_vector_type(8)));
typedef unsigned int u32x4 __attribute__((ext_vector_type(4)));
typedef int i32x4 __attribute__((ext_vector_type(4)));
typedef int i32x8 __attribute__((ext_vector_type(8)));

#define TPB   256
#define WAVES 8
#define SUBT  4          // 16-wide N sub-tiles per wave
#define BN    512        // block N tile = 8 waves * 64 cols
#define KC    16         // K chunk staged per TDM transfer
#define MAXK  200
#define NCLS  1000

// ---------------------------------------------------------------------------
// TDM: async 2-D tile load (rows x tile_cols of fp32) from global into LDS.
// D# group0: count=1 | lds_addr | global_addr | type=2 (bits 127:126).
// D# group1: data_size=4B | tensor_dim0=cols_avail (zero-fill OOB edge) |
//            tensor_dim1=rows | tile_dim0=tile_cols | tile_dim1=rows |
//            tensor_dim0_stride=row_stride (elements).
// ---------------------------------------------------------------------------
__device__ __forceinline__ void tdm_load_2d(unsigned lds_off, const float* gptr,
                                            unsigned rows, unsigned cols_avail,
                                            unsigned tile_cols,
                                            unsigned row_stride_elems) {
  unsigned long long ga = (unsigned long long)gptr;
  u32x4 g0;
  g0[0] = 1u;                                              // count=1, user D#
  g0[1] = lds_off;                                         // lds_addr (bytes)
  g0[2] = (unsigned)(ga & 0xFFFFFFFFull);                  // global_addr lo
  g0[3] = (unsigned)((ga >> 32) & 0x01FFFFFFull) | 0x80000000u;  // hi | type=2
  i32x8 g1;
  g1[0] = (int)(2u << 16);                                 // data_size = 4B
  g1[1] = (int)((cols_avail & 0xFFFFu) << 16);             // dim0[15:0]
  g1[2] = (int)((cols_avail >> 16) | ((rows & 0xFFFFu) << 16)); // dim0 hi|dim1 lo
  g1[3] = (int)((rows >> 16) | (tile_cols << 16));         // dim1 hi|tile_dim0
  g1[4] = (int)(rows & 0xFFFFu);                           // tile_dim1, tile_dim2=0
  g1[5] = (int)row_stride_elems;                           // dim0_stride lo
  g1[6] = 0;                                               // stride hi | dim1_stride
  g1[7] = 0;
  i32x4 z4 = {0, 0, 0, 0};
#if __clang_major__ >= 23
  i32x8 z8 = {0, 0, 0, 0, 0, 0, 0, 0};
  __builtin_amdgcn_tensor_load_to_lds(g0, g1, z4, z4, z8, 0);
#else
  __builtin_amdgcn_tensor_load_to_lds(g0, g1, z4, z4, 0);
#endif
}

// ---------------------------------------------------------------------------
// Pass 1: sim = A(MxK) * B(KxN), fp32 WMMA 16x16x4.
// Block tile: 16(M) x 512(N). TDM double-buffers 16x512 B panels into LDS;
// wave 0 issues/waits (TENSORcnt), __syncthreads fences consumers.
// B fragments come from direct __shared__ indexing -> ds_load_b32 (DScnt only).
// ---------------------------------------------------------------------------
__global__ __launch_bounds__(TPB) void gemm_knn_wmma(
    const float* __restrict__ A, const float* __restrict__ Bm,
    float* __restrict__ C, int M, int N, int K) {
  __shared__ float lds_B[2 * KC][BN];  // 64 KB double buffer (flattened)

  const int lane = threadIdx.x & 31;
  const int wave = threadIdx.x >> 5;
  const int half = lane >> 4;          // K sub-offset {0,1} vs {2,3}
  const int l15  = lane & 15;
  const int m0   = blockIdx.x * 16;
  const int nblk = blockIdx.y * BN;    // block's first column

  const int NC      = K / KC;          // K chunks
  const int colBase = wave * 64 + l15; // this lane's column within the panel

  // per-lane A pointer (row m0+l15, K offset 2*half), advanced KC per chunk
  const float* Ap = A + (size_t)(m0 + l15) * K + 2 * half;

  const unsigned colsAvail = (unsigned)(N - nblk);   // zero-fill edge via D#

  // preload chunk 0 into buffer 0
  if (threadIdx.x < 32) {
    tdm_load_2d((unsigned)(size_t)&lds_B[0][0],
                Bm + nblk, KC, colsAvail, BN, (unsigned)N);
  }

  v8f acc[SUBT] = {};

  for (int ic = 0; ic < NC; ++ic) {
    const int cur = ic & 1;
    if (threadIdx.x < 32) {
      if (ic + 1 < NC) {
        tdm_load_2d((unsigned)(size_t)&lds_B[(cur ^ 1) * KC][0],
                    Bm + (size_t)(ic + 1) * KC * N + nblk,
                    KC, colsAvail, BN, (unsigned)N);
        __builtin_amdgcn_s_wait_tensorcnt((short)1);  // chunk ic done
      } else {
        __builtin_amdgcn_s_wait_tensorcnt((short)0);
      }
    }
    __syncthreads();                    // panel `cur` visible to all waves

    const int rbase = cur * KC + 2 * half;   // runtime part of the row index
#pragma unroll
    for (int ks = 0; ks < KC / 4; ++ks) {
      v2f a;
      a.x = Ap[4 * ks];
      a.y = Ap[4 * ks + 1];
#pragma unroll
      for (int t = 0; t < SUBT; ++t) {
        v2f b;
        b.x = lds_B[rbase + 4 * ks + 0][colBase + t * 16];  // ds_load_b32
        b.y = lds_B[rbase + 4 * ks + 1][colBase + t * 16];  // ds_load_b32
        acc[t] = __builtin_amdgcn_wmma_f32_16x16x4_f32(
            false, a, false, b, (short)0, acc[t], false, false);
      }
    }
    Ap += KC;
    __syncthreads();                    // done reading before buffer reuse
  }

  // Store: D layout -> VGPR r holds M = r + 8*half, N = lane&15.
#pragma unroll
  for (int t = 0; t < SUBT; ++t) {
    long col = (long)nblk + colBase + t * 16;
    if (col < (long)N) {
#pragma unroll
      for (int r = 0; r < 8; ++r)
        C[(size_t)(m0 + r + 8 * half) * N + col] = acc[t][r];
    }
  }
}

// ---------------------------------------------------------------------------
// Pass 2: per-row exact top-200 via MSB radix select, bitonic sort, softmax,
// class scatter for k in {10,20,100,200}. One block per row, deterministic.
// ---------------------------------------------------------------------------
__device__ __forceinline__ unsigned fkey(float f) {
  unsigned u = __float_as_uint(f);
  return (u & 0x80000000u) ? ~u : (u | 0x80000000u);  // order-preserving
}

__global__ __launch_bounds__(TPB) void knn_topk_scatter(
    const float* __restrict__ sim, const int* __restrict__ labels,
    float* __restrict__ out, int N, int Brows) {
  const int tid = threadIdx.x;
  const int b   = blockIdx.x;
  const size_t rowbase = (size_t)b * (size_t)N;

  __shared__ unsigned hist[256];
  __shared__ float    svals[256];
  __shared__ int      sidx[256];
  __shared__ float    red[256];
  __shared__ float    prob[4 * NCLS];
  __shared__ unsigned s_thresh;
  __shared__ int      s_rem;
  __shared__ unsigned cnt_hi, cnt_eq;

  // ---- radix select the key of the 200th-largest element ----
  unsigned prefix = 0;
  int rem = MAXK;
  for (int pass = 0; pass < 4; ++pass) {
    const int shift = 24 - 8 * pass;
    hist[tid] = 0;
    __syncthreads();
    const unsigned pmask = (pass == 0) ? 0u : (0xFFFFFFFFu << (shift + 8));
    for (int i = tid; i < N; i += TPB) {
      unsigned key = fkey(sim[rowbase + i]);
      if ((key & pmask) == prefix)
        atomicAdd(&hist[(key >> shift) & 255u], 1u);
    }
    __syncthreads();
    if (tid == 0) {
      int cum = 0, bin = 0;
      for (int d = 255; d >= 0; --d) {
        int c = (int)hist[d];
        if (cum + c >= rem) { bin = d; break; }
        cum += c;
      }
      s_thresh = prefix | ((unsigned)bin << shift);
      s_rem    = rem - cum;
    }
    __syncthreads();
    prefix = s_thresh;
    rem    = s_rem;
    __syncthreads();
  }
  const unsigned thresh = prefix;   // count(key > thresh) = 200 - rem

  // ---- collect exactly 200 (value,index) pairs ----
  if (tid == 0) { cnt_hi = 0; cnt_eq = 0; }
  svals[tid] = -__builtin_inff();
  sidx[tid]  = 0x7FFFFFFF;
  __syncthreads();
  for (int i = tid; i < N; i += TPB) {
    float v = sim[rowbase + i];
    unsigned key = fkey(v);
    if (key > thresh) {
      unsigned p = atomicAdd(&cnt_hi, 1u);
      svals[p] = v; sidx[p] = i;
    } else if (key == thresh) {
      unsigned p = atomicAdd(&cnt_eq, 1u);
      if (p < (unsigned)rem) { svals[255 - p] = v; sidx[255 - p] = i; }
    }
  }
  __syncthreads();

  // ---- bitonic sort, descending by value (ties: lower index first) ----
  for (int kk = 2; kk <= 256; kk <<= 1) {
    for (int j = kk >> 1; j > 0; j >>= 1) {
      __syncthreads();
      int ixj = tid ^ j;
      if (ixj > tid) {
        float v0 = svals[tid], v1 = svals[ixj];
        int   i0 = sidx[tid],  i1 = sidx[ixj];
        bool desc = ((tid & kk) == 0);
        bool gt = (v1 > v0) || (v1 == v0 && i1 < i0);
        if (gt == desc) {
          svals[tid] = v1; svals[ixj] = v0;
          sidx[tid]  = i1; sidx[ixj]  = i0;
        }
      }
    }
  }
  __syncthreads();

  // ---- softmax over top-200 with T = 0.07 ----
  const float vmax = svals[0];
  float e = 0.0f;
  if (tid < MAXK) e = __expf((svals[tid] - vmax) * (1.0f / 0.07f));
  red[tid] = e;
  __syncthreads();
  for (int s = TPB >> 1; s > 0; s >>= 1) {
    if (tid < s) red[tid] += red[tid + s];
    __syncthreads();
  }
  const float inv = 1.0f / red[0];
  __syncthreads();
  if (tid < MAXK) {
    svals[tid] = e * inv;                       // weight
    hist[tid]  = (unsigned)labels[sidx[tid]];   // neighbor label
  }
  for (int i = tid; i < 4 * NCLS; i += TPB) prob[i] = 0.0f;
  __syncthreads();

  // ---- deterministic serial scatter (200 items, trivial cost) ----
  if (tid == 0) {
    for (int j = 0; j < MAXK; ++j) {
      int   lab = (int)hist[j];
      float w   = svals[j];
      prob[3 * NCLS + lab] += w;            // k = 200
      if (j < 100) prob[2 * NCLS + lab] += w;
      if (j < 20)  prob[1 * NCLS + lab] += w;
      if (j < 10)  prob[0 * NCLS + lab] += w;
    }
  }
  __syncthreads();

  const size_t plane = (size_t)Brows * NCLS;
  const size_t obase = (size_t)b * NCLS;
  for (int c = tid; c < NCLS; c += TPB) {
    out[0 * plane + obase + c] = prob[0 * NCLS + c];
    out[1 * plane + obase + c] = prob[1 * NCLS + c];
    out[2 * plane + obase + c] = prob[2 * NCLS + c];
    out[3 * plane + obase + c] = prob[3 * NCLS + c];
  }
}

// ---------------------------------------------------------------------------
extern "C" void kernel_launch(void* const* d_in, const int* in_sizes, int n_in,
                              void* d_out, int out_size, void* d_ws, size_t ws_size,
                              hipStream_t stream) {
  const float* A      = (const float*)d_in[0];   // [B, D]
  const float* Bm     = (const float*)d_in[1];   // [D, N]
  const int*   labels = (const int*)d_in[2];     // [N]
  float*       out    = (float*)d_out;           // 4 x [B, 1000] concat

  const int N = in_sizes[2];
  const int D = in_sizes[1] / N;
  const int Brows = in_sizes[0] / D;

  float* sim = (float*)d_ws;                     // [B, N] fp32 scratch

  dim3 g1(Brows / 16, (N + BN - 1) / BN);
  gemm_knn_wmma<<<g1, TPB, 0, stream>>>(A, Bm, sim, Brows, N, D);

  knn_topk_scatter<<<Brows, TPB, 0, stream>>>(sim, labels, out, N, Brows);
}